// Gemma3nAttention_4990751998314
// MI455X (gfx1250) — compile-verified
//
#include <hip/hip_runtime.h>
#include <hip/hip_bf16.h>

typedef __attribute__((ext_vector_type(2))) float v2f;
typedef __attribute__((ext_vector_type(8))) float v8f;
typedef __attribute__((ext_vector_type(4))) unsigned int u32x4;
typedef __attribute__((ext_vector_type(8))) int i32x8;
typedef __attribute__((ext_vector_type(4))) int i32x4;

#define HIDDEN    2048
#define NUM_HEADS 8
#define NUM_KV    2
#define HEAD_DIM  256
#define SEQ       2048
#define BATCH     2
#define QKV_OUT   3072   // (8 + 2*2) * 256
#define Q_SIZE    2048
#define KV_SIZE   512

// ---------------------------------------------------------------------------
// V_WMMA_F32_16X16X4_F32 fragment helper (D = A(16x4) * B(4x16) + C)
// A: lane l -> row m = l&15 ; VGPR v holds K = (l>=16 ? 2 : 0) + v
// B: lane l -> col n = l&15 ; VGPR v holds K = (l>=16 ? 2 : 0) + v
// C/D: lane l -> col n = l&15 ; VGPR v holds row m = v + (l>=16 ? 8 : 0)
// ---------------------------------------------------------------------------
__device__ __forceinline__ v8f wmma_f32(v2f a, v2f b, v8f c) {
    return __builtin_amdgcn_wmma_f32_16x16x4_f32(false, a, false, b,
                                                 (short)0, c, false, false);
}

// ---------------------------------------------------------------------------
// TDM: DMA a 2-D fp32 tile (tile_w x tile_h elems, row stride `row_stride`
// elems in global) into LDS at byte offset lds_addr, inserting `pad_amt+1`
// dwords of LDS padding every 2^(pad_int+1) dwords (hardware bank-spread).
// Descriptor packing per CDNA5 ISA (groups 2/3 and the
// trailing group = 0 for 2-D tiles). This toolchain uses the 6-arg builtin:
// (u32x4 g0, i32x8 g1, i32x4, i32x4, i32x8, i32 cpol).
// Issue from ONE wave only; complete via s_wait_tensorcnt.
// ---------------------------------------------------------------------------
__device__ __forceinline__ void tdm_load_2d(unsigned lds_addr,
                                            const float* gsrc,
                                            unsigned tile_w, unsigned tile_h,
                                            unsigned row_stride,
                                            unsigned pad_int, unsigned pad_amt,
                                            bool pad_en) {
    const unsigned long long ga = (unsigned long long)(const void*)gsrc;
    u32x4 g0;
    g0.x = 1u;                                   // count=1, user-mode load
    g0.y = lds_addr;                             // LDS byte address
    g0.z = (unsigned)(ga & 0xffffffffull);       // global_addr[31:0]
    g0.w = (unsigned)((ga >> 32) & 0x1ffffffull) // global_addr[56:32]
           | (2u << 30);                         // type = 2 ("image")
    i32x8 g1;
    unsigned d0 = (2u << 16);                    // data_size = 4 bytes
    if (pad_en)
        d0 |= (1u << 20) | (pad_int << 22) | (pad_amt << 25);
    const unsigned td0 = row_stride;             // tensor_dim0 (OOB bound)
    const unsigned td1 = 1u << 30;               // tensor_dim1: huge
    g1[0] = (int)d0;
    g1[1] = (int)((td0 & 0xffffu) << 16);        // [63:48] = td0 lo16
    g1[2] = (int)((td0 >> 16) | ((td1 & 0xffffu) << 16));
    g1[3] = (int)((td1 >> 16) | (tile_w << 16)); // [127:112] = tile_dim0
    g1[4] = (int)tile_h;                         // tile_dim1; tile_dim2 = 0
    g1[5] = (int)row_stride;                     // tensor_dim0_stride[31:0]
    g1[6] = 0;                                   // stride0 hi / stride1 lo
    g1[7] = 0;
    const i32x4 z4 = {0, 0, 0, 0};
    const i32x8 z8 = {0, 0, 0, 0, 0, 0, 0, 0};
    __builtin_amdgcn_tensor_load_to_lds(g0, g1, z4, z4, z8, 0);
}

__device__ __forceinline__ unsigned lds_off(const void* p) {
    return (unsigned)(unsigned long long)p;      // flat LDS ptr -> low 32 bits
}

// ===========================================================================
// Kernels 1 & 4: tiled fp32 WMMA GEMM, TDM-staged + double-buffered K panels.
// block = 128 threads (4 waves); block tile 128x64; each wave owns 32x64.
// ===========================================================================
__global__ __launch_bounds__(128)
void gemm_f32_wmma(const float* __restrict__ A, const float* __restrict__ B,
                   float* __restrict__ C, int M, int N, int K) {
    __shared__ float As[2][128][18];   // stride 18: 8B-aligned float2 rows
    __shared__ float Bs[2][16][65];

    const int tid  = threadIdx.x;
    const int w    = tid >> 5;
    const int lane = tid & 31;
    const int hx   = lane >> 4;
    const int l16  = lane & 15;

    const int m0 = blockIdx.y * 128;
    const int n0 = blockIdx.x * 64;
    const int nPanels = K >> 4;

    v8f acc[2][4];
    #pragma unroll
    for (int mt = 0; mt < 2; ++mt)
        #pragma unroll
        for (int nt = 0; nt < 4; ++nt) acc[mt][nt] = v8f{};

    if (w == 0) {                       // prologue: stage panel 0 -> buf 0
        tdm_load_2d(lds_off(&As[0][0][0]), A + (size_t)m0 * K,
                    16, 128, (unsigned)K, 3, 1, true);   // pad 16 -> 18
        tdm_load_2d(lds_off(&Bs[0][0][0]), B + (size_t)n0,
                    64, 16, (unsigned)N, 5, 0, true);    // pad 64 -> 65
    }

    for (int p = 0; p < nPanels; ++p) {
        const int buf = p & 1;
        if (w == 0) {
            if (p + 1 < nPanels) {      // stage next panel into other buffer
                const int k1 = (p + 1) << 4;
                tdm_load_2d(lds_off(&As[buf ^ 1][0][0]),
                            A + (size_t)m0 * K + k1, 16, 128, (unsigned)K,
                            3, 1, true);
                tdm_load_2d(lds_off(&Bs[buf ^ 1][0][0]),
                            B + (size_t)k1 * N + n0, 64, 16, (unsigned)N,
                            5, 0, true);
                __builtin_amdgcn_s_wait_tensorcnt((short)2);  // panel p done
            } else {
                __builtin_amdgcn_s_wait_tensorcnt((short)0);
            }
        }
        __syncthreads();

        #pragma unroll
        for (int kk = 0; kk < 4; ++kk) {
            v2f a[2];
            #pragma unroll
            for (int mt = 0; mt < 2; ++mt) {
                const float2 t = *(const float2*)(
                    &As[buf][w * 32 + mt * 16 + l16][kk * 4 + hx * 2]);
                a[mt].x = t.x; a[mt].y = t.y;
            }
            #pragma unroll
            for (int nt = 0; nt < 4; ++nt) {
                v2f b;
                b.x = Bs[buf][kk * 4 + hx * 2    ][nt * 16 + l16];
                b.y = Bs[buf][kk * 4 + hx * 2 + 1][nt * 16 + l16];
                acc[0][nt] = wmma_f32(a[0], b, acc[0][nt]);
                acc[1][nt] = wmma_f32(a[1], b, acc[1][nt]);
            }
        }
        __syncthreads();
    }

    #pragma unroll
    for (int mt = 0; mt < 2; ++mt)
        #pragma unroll
        for (int nt = 0; nt < 4; ++nt)
            #pragma unroll
            for (int v = 0; v < 8; ++v)
                C[(size_t)(m0 + w * 32 + mt * 16 + hx * 8 + v) * N +
                  n0 + nt * 16 + l16] = acc[mt][nt][v];
}

// ===========================================================================
// Kernel 2: per-head RMSNorm + RoPE, scatter into (b, head, s, d) layouts
// ===========================================================================
__global__ __launch_bounds__(256)
void normrope_kernel(const float* __restrict__ qkv,
                     const int* __restrict__ positions,
                     const float* __restrict__ qw, const float* __restrict__ kw,
                     float* __restrict__ Qb, float* __restrict__ Kb,
                     float* __restrict__ Vb) {
    const int s    = blockIdx.x;
    const int b    = blockIdx.y;
    const int w    = threadIdx.x >> 5;
    const int lane = threadIdx.x & 31;

    const float* base = qkv + (size_t)(b * SEQ + s) * QKV_OUT;
    const float  pos  = (float)positions[b * SEQ + s];

    for (int hh = w; hh < 12; hh += 8) {     // 8 q heads + 2 k + 2 v
        int col0;
        if (hh < 8)       col0 = hh * HEAD_DIM;
        else if (hh < 10) col0 = Q_SIZE + (hh - 8) * HEAD_DIM;
        else              col0 = Q_SIZE + KV_SIZE + (hh - 10) * HEAD_DIM;
        const float* xp = base + col0;

        float x[8];
        float ss = 0.0f;
        #pragma unroll
        for (int r = 0; r < 8; ++r) {
            x[r] = xp[lane + 32 * r];
            ss += x[r] * x[r];
        }
        #pragma unroll
        for (int off = 16; off >= 1; off >>= 1)
            ss += __shfl_xor(ss, off, 32);
        const float inv = rsqrtf(ss * (1.0f / 256.0f) + 1e-6f);

        #pragma unroll
        for (int r = 0; r < 8; ++r) {
            const int d = lane + 32 * r;
            const float wgt = (hh < 8) ? qw[d] : (hh < 10 ? kw[d] : 1.0f);
            x[r] *= inv * wgt;
        }

        float y[8];
        if (hh < 10) {  // RoPE (NeoX split-half): pair (d, d+128) same lane
            #pragma unroll
            for (int r = 0; r < 4; ++r) {
                const int p = lane + 32 * r;
                const float freq =
                    pos * __powf(1.0e6f, -(float)p * (1.0f / 128.0f));
                float sn, cs;
                __sincosf(freq, &sn, &cs);
                y[r]     = x[r]     * cs - x[r + 4] * sn;
                y[r + 4] = x[r + 4] * cs + x[r]     * sn;
            }
        } else {
            #pragma unroll
            for (int r = 0; r < 8; ++r) y[r] = x[r];
        }

        float* dst;
        if (hh < 8)
            dst = Qb + ((size_t)(b * NUM_HEADS + hh) * SEQ + s) * HEAD_DIM;
        else if (hh < 10)
            dst = Kb + ((size_t)(b * NUM_KV + (hh - 8)) * SEQ + s) * HEAD_DIM;
        else
            dst = Vb + ((size_t)(b * NUM_KV + (hh - 10)) * SEQ + s) * HEAD_DIM;
        #pragma unroll
        for (int r = 0; r < 8; ++r) dst[lane + 32 * r] = y[r];
    }
}

// ===========================================================================
// Kernel 3: causal flash attention, fp32 WMMA, TDM-staged K/V tiles
// grid (S/16/8, H, B), block 256 = 8 waves, wave w -> q-tile blockIdx.x*8+w
// ===========================================================================
__global__ __launch_bounds__(256)
void attn_kernel(const float* __restrict__ Qb, const float* __restrict__ Kb,
                 const float* __restrict__ Vb, float* __restrict__ Ob) {
    __shared__ float Ks[16][260];      // stride 260 via TDM pad (256 + 4)
    __shared__ float Vs[16][260];
    __shared__ float Pw[8][16][18];    // per-wave P transpose buffer

    const int tid  = threadIdx.x;
    const int w    = tid >> 5;
    const int lane = tid & 31;
    const int hx   = lane >> 4;
    const int l16  = lane & 15;

    const int b     = blockIdx.z;
    const int h     = blockIdx.y;
    const int kvh   = h >> 2;          // GQA group = 4
    const int qtile = blockIdx.x * 8 + w;
    const int jmax  = blockIdx.x * 8 + 7;

    // Q tile (16x256) -> A-layout fragments in registers
    const float* Qp =
        Qb + ((size_t)(b * NUM_HEADS + h) * SEQ + qtile * 16 + l16) * HEAD_DIM;
    v2f qa[64];
    #pragma unroll
    for (int kk = 0; kk < 64; ++kk) {
        const float2 t = *(const float2*)(Qp + kk * 4 + hx * 2);
        qa[kk].x = t.x; qa[kk].y = t.y;
    }

    v8f o[16];
    #pragma unroll
    for (int nt = 0; nt < 16; ++nt) o[nt] = v8f{};
    float mrow[8], lrow[8];
    #pragma unroll
    for (int v = 0; v < 8; ++v) { mrow[v] = -3.0e38f; lrow[v] = 0.0f; }

    const float* Kbase = Kb + (size_t)(b * NUM_KV + kvh) * SEQ * HEAD_DIM;
    const float* Vbase = Vb + (size_t)(b * NUM_KV + kvh) * SEQ * HEAD_DIM;

    for (int j = 0; j <= jmax; ++j) {
        if (w == 0) {                  // TDM-stage this 16x256 key/value tile
            tdm_load_2d(lds_off(&Ks[0][0]),
                        Kbase + (size_t)j * 16 * HEAD_DIM, 256, 16, 256,
                        7, 3, true);   // pad 256 -> 260
            tdm_load_2d(lds_off(&Vs[0][0]),
                        Vbase + (size_t)j * 16 * HEAD_DIM, 256, 16, 256,
                        7, 3, true);
            __builtin_amdgcn_s_wait_tensorcnt((short)0);
        }
        __syncthreads();

        if (j <= qtile) {
            // S = Q * K^T : 64 WMMAs over D=256
            v8f sc = v8f{};
            #pragma unroll
            for (int kk = 0; kk < 64; ++kk) {
                v2f bfrag;
                const float2 t = *(const float2*)(&Ks[l16][kk * 4 + hx * 2]);
                bfrag.x = t.x; bfrag.y = t.y;
                sc = wmma_f32(qa[kk], bfrag, sc);
            }

            // Online softmax in C/D layout (row = hx*8+v, col = l16)
            const bool diag = (j == qtile);
            #pragma unroll
            for (int v = 0; v < 8; ++v) {
                const int m = hx * 8 + v;
                float sv = sc[v];
                if (diag && (l16 > m)) sv = -3.0e38f;
                float rmax = sv;
                #pragma unroll
                for (int off = 8; off >= 1; off >>= 1)
                    rmax = fmaxf(rmax, __shfl_xor(rmax, off, 16));
                const float mnew = fmaxf(mrow[v], rmax);
                const float p    = __expf(sv - mnew);
                float rsum = p;
                #pragma unroll
                for (int off = 8; off >= 1; off >>= 1)
                    rsum += __shfl_xor(rsum, off, 16);
                const float corr = __expf(mrow[v] - mnew);
                lrow[v] = lrow[v] * corr + rsum;
                mrow[v] = mnew;
                #pragma unroll
                for (int nt = 0; nt < 16; ++nt) o[nt][v] *= corr;
                Pw[w][m][l16] = p;     // D-layout -> LDS for A-layout reload
            }

            // O += P * V : 4 WMMAs (K=16 keys) per 16-col tile of D
            #pragma unroll
            for (int kk2 = 0; kk2 < 4; ++kk2) {
                v2f a2;
                const float2 t =
                    *(const float2*)(&Pw[w][l16][kk2 * 4 + hx * 2]);
                a2.x = t.x; a2.y = t.y;
                #pragma unroll
                for (int nt = 0; nt < 16; ++nt) {
                    v2f b2;
                    b2.x = Vs[kk2 * 4 + hx * 2    ][nt * 16 + l16];
                    b2.y = Vs[kk2 * 4 + hx * 2 + 1][nt * 16 + l16];
                    o[nt] = wmma_f32(a2, b2, o[nt]);
                }
            }
        }
        __syncthreads();
    }

    // Normalize, scatter into (b, s, h*256 + d)
    #pragma unroll
    for (int v = 0; v < 8; ++v) {
        const float inv = 1.0f / lrow[v];
        const int q = qtile * 16 + hx * 8 + v;
        float* row = Ob + (size_t)(b * SEQ + q) * (NUM_HEADS * HEAD_DIM) +
                     h * HEAD_DIM;
        #pragma unroll
        for (int nt = 0; nt < 16; ++nt)
            row[nt * 16 + l16] = o[nt][v] * inv;
    }
}

// ===========================================================================
extern "C" void kernel_launch(void* const* d_in, const int* in_sizes, int n_in,
                              void* d_out, int out_size, void* d_ws,
                              size_t ws_size, hipStream_t stream) {
    const float* hidden    = (const float*)d_in[0];
    const int*   positions = (const int*)d_in[1];
    const float* w_qkv     = (const float*)d_in[2];
    const float* w_o       = (const float*)d_in[3];
    const float* q_norm_w  = (const float*)d_in[4];
    const float* k_norm_w  = (const float*)d_in[5];
    float*       out       = (float*)d_out;

    float* ws = (float*)d_ws;
    const size_t rows = (size_t)BATCH * SEQ;               // 4096
    float* qkv  = ws;                                      // 4096*3072
    float* attn = ws;                                      // reuses qkv region
    float* Qb   = ws + rows * QKV_OUT;
    float* Kb   = Qb + (size_t)BATCH * NUM_HEADS * SEQ * HEAD_DIM;
    float* Vb   = Kb + (size_t)BATCH * NUM_KV * SEQ * HEAD_DIM;

    {   // 1) QKV projection: (4096x2048) * (2048x3072)
        dim3 grid(QKV_OUT / 64, rows / 128);
        gemm_f32_wmma<<<grid, 128, 0, stream>>>(hidden, w_qkv, qkv,
                                                (int)rows, QKV_OUT, HIDDEN);
    }
    {   // 2) RMSNorm + RoPE
        dim3 grid(SEQ, BATCH);
        normrope_kernel<<<grid, 256, 0, stream>>>(qkv, positions, q_norm_w,
                                                  k_norm_w, Qb, Kb, Vb);
    }
    {   // 3) Causal flash attention (overwrites qkv region)
        dim3 grid(SEQ / 16 / 8, NUM_HEADS, BATCH);
        attn_kernel<<<grid, 256, 0, stream>>>(Qb, Kb, Vb, attn);
    }
    {   // 4) Output projection: (4096x2048) * (2048x2048)
        dim3 grid(HIDDEN / 64, rows / 128);
        gemm_f32_wmma<<<grid, 128, 0, stream>>>(attn, w_o, out, (int)rows,
                                                HIDDEN, HIDDEN);
    }
}